// GumbelTopKRouter_19215683682901
// MI455X (gfx1250) — compile-verified
//
#include <hip/hip_runtime.h>
#include <stdint.h>

// ---------------------------------------------------------------------------
// Gumbel-softmax top-k router, forward only.
//   B=2048 rows, N=16384 cols, k=24, T=0.7, eps=1e-10
// Streaming row reduction: 256 MB read once -> ~11us floor @ 23.3 TB/s, with
// ~3 single-instruction transcendentals/element (2x v_log_f32 + 1x v_exp_f32)
// pacing alongside memory. Data path: per-lane global_load_async_to_lds_b128
// in SADDR (GVS) form with NT hint, 4-slot ring, throttled by s_wait_asynccnt.
// Each thread consumes exactly the LDS bytes it staged -> no barriers in the
// streaming loop (ASYNCcnt is per-wave).
// ---------------------------------------------------------------------------

#define N_COLS   16384
#define B_ROWS   2048
#define THREADS  512
#define NWAVES   (THREADS / 32)         // 16
#define TILE     2048                   // floats per tile per array (= THREADS*4)
#define NTILES   (N_COLS / TILE)        // 8
#define SLOTS    4                      // ring depth (4 slots x 2 arrays = 64 KB LDS)
#define ZPT      (N_COLS / THREADS)     // 32 z-values per thread
#define KSEL     24

#define LN2f     0.69314718055994531f
// (1/0.7) * log2(e)
#define C_EXP2   2.0609929155f
#define EPSF     1e-10f

__device__ __forceinline__ float fastlog2(float x) {
#if __has_builtin(__builtin_amdgcn_logf)
  return __builtin_amdgcn_logf(x);        // v_log_f32
#else
  return __log2f(x);
#endif
}

__device__ __forceinline__ float fastexp2(float x) {
#if __has_builtin(__builtin_amdgcn_exp2f)
  return __builtin_amdgcn_exp2f(x);       // v_exp_f32
#else
  return exp2f(x);
#endif
}

// z = logit + gumbel(u);  gumbel = -ln(-ln(u+eps)+eps), ln via log2 * ln2
__device__ __forceinline__ float gumbel_z(float lg, float u) {
  float inner = -LN2f * fastlog2(u + EPSF);
  float g     = -LN2f * fastlog2(inner + EPSF);
  return lg + g;
}

// async global -> LDS (GVS form: SGPR64 base + VGPR32 byte offset), per-lane
// 16B, non-temporal streaming hint; tracked by ASYNCcnt, no VGPR writeback.
#define ASYNC_LD_B128(ldsoff, voff, sbase)                                    \
  asm volatile("global_load_async_to_lds_b128 %0, %1, %2 th:TH_LOAD_NT"       \
               :: "v"(ldsoff), "v"(voff), "s"(sbase) : "memory")

#define WAIT_ASYNC(n)                                                         \
  asm volatile("s_wait_asynccnt " #n ::: "memory")

// one tile = 1 async b128 per thread per array (2 ASYNCcnt increments);
// the same global byte offset serves both arrays (different SGPR bases).
#define ISSUE_TILE(t, slot) do {                                              \
    unsigned _go = go_base + (unsigned)((t) * TILE * 4);                      \
    unsigned _dl = dl_base + (unsigned)((slot) * TILE * 4);                   \
    unsigned _du = du_base + (unsigned)((slot) * TILE * 4);                   \
    ASYNC_LD_B128(_dl, _go, Lrow);                                            \
    ASYNC_LD_B128(_du, _go, Urow);                                            \
  } while (0)

// consume the 4 elements this thread staged for tile t (constant indices so
// z[] stays in VGPRs)
#define CONSUME_TILE(t, slot) do {                                            \
    const float4 lv = *(const float4*)(&sL[(slot) * TILE + (tid << 2)]);      \
    const float4 uv = *(const float4*)(&sU[(slot) * TILE + (tid << 2)]);      \
    const int cbase = (t) * TILE + (tid << 2);                                \
    z[(t)*4+0] = gumbel_z(lv.x, uv.x);                                        \
    if (z[(t)*4+0] > m) { m = z[(t)*4+0]; am = cbase + 0; }                   \
    z[(t)*4+1] = gumbel_z(lv.y, uv.y);                                        \
    if (z[(t)*4+1] > m) { m = z[(t)*4+1]; am = cbase + 1; }                   \
    z[(t)*4+2] = gumbel_z(lv.z, uv.z);                                        \
    if (z[(t)*4+2] > m) { m = z[(t)*4+2]; am = cbase + 2; }                   \
    z[(t)*4+3] = gumbel_z(lv.w, uv.w);                                        \
    if (z[(t)*4+3] > m) { m = z[(t)*4+3]; am = cbase + 3; }                   \
  } while (0)

__global__ __launch_bounds__(THREADS)
void router_kernel(const float* __restrict__ logits,
                   const float* __restrict__ noise,
                   float* __restrict__ out,
                   float* __restrict__ ws) {
  __shared__ __align__(16) float sL[SLOTS * TILE];   // 32 KB, 4-slot ring
  __shared__ __align__(16) float sU[SLOTS * TILE];   // 32 KB
  __shared__ float redm[NWAVES];
  __shared__ int   reda[NWAVES];
  __shared__ float reds[NWAVES];
  __shared__ float bmax_sh;
  __shared__ int   barg_sh;

  const int tid = (int)threadIdx.x;
  const int row = (int)blockIdx.x;
  const float* Lrow = logits + (size_t)row * N_COLS;
  const float* Urow = noise  + (size_t)row * N_COLS;

  // low 32 bits of the flat address of a __shared__ object = LDS byte offset
  const unsigned dl_base = (unsigned)(uintptr_t)&sL[0] + (unsigned)(tid << 4);
  const unsigned du_base = (unsigned)(uintptr_t)&sU[0] + (unsigned)(tid << 4);
  const unsigned go_base = (unsigned)(tid << 4);

  float z[ZPT];
  float m  = -__builtin_inff();
  int   am = 0;

  // prologue: fill 4-slot pipeline (8 async ops outstanding)
  ISSUE_TILE(0, 0); ISSUE_TILE(1, 1); ISSUE_TILE(2, 2); ISSUE_TILE(3, 3);

  // steady state: 3 newer tiles (6 ops) may remain outstanding
  WAIT_ASYNC(6); CONSUME_TILE(0, 0); ISSUE_TILE(4, 0);
  WAIT_ASYNC(6); CONSUME_TILE(1, 1); ISSUE_TILE(5, 1);
  WAIT_ASYNC(6); CONSUME_TILE(2, 2); ISSUE_TILE(6, 2);
  WAIT_ASYNC(6); CONSUME_TILE(3, 3); ISSUE_TILE(7, 3);
  WAIT_ASYNC(6); CONSUME_TILE(4, 0);
  WAIT_ASYNC(4); CONSUME_TILE(5, 1);
  WAIT_ASYNC(2); CONSUME_TILE(6, 2);
  WAIT_ASYNC(0); CONSUME_TILE(7, 3);

  // ---- block argmax (value desc, index asc on ties -> matches jnp.argmax)
  const int lane = tid & 31;
  const int wave = tid >> 5;
#pragma unroll
  for (int off = 16; off > 0; off >>= 1) {
    float om = __shfl_xor(m, off, 32);
    int   oa = __shfl_xor(am, off, 32);
    if (om > m || (om == m && oa < am)) { m = om; am = oa; }
  }
  if (lane == 0) { redm[wave] = m; reda[wave] = am; }
  __syncthreads();
  if (tid == 0) {
    float bm = redm[0]; int ba = reda[0];
#pragma unroll
    for (int w = 1; w < NWAVES; ++w) {
      float om = redm[w]; int oa = reda[w];
      if (om > bm || (om == bm && oa < ba)) { bm = om; ba = oa; }
    }
    bmax_sh = bm; barg_sh = ba;
  }
  __syncthreads();
  const float bm = bmax_sh;

  // ---- pass B: sum exp((z - max)/T) over the row (z stays in VGPRs)
  float ssum = 0.0f;
#pragma unroll
  for (int i = 0; i < ZPT; ++i)
    ssum += fastexp2((z[i] - bm) * C_EXP2);
#pragma unroll
  for (int off = 16; off > 0; off >>= 1)
    ssum += __shfl_xor(ssum, off, 32);
  if (lane == 0) reds[wave] = ssum;
  __syncthreads();

  if (tid == 0) {
    float Zs = 0.0f;
#pragma unroll
    for (int w = 0; w < NWAVES; ++w) Zs += reds[w];
    const float s = 1.0f / Zs;                 // softmax value at argmax
    const float v = (1.0f - s) + s;            // straight-through numerics
    const int   a = barg_sh;

    float* idxOut = out + (size_t)row * KSEL;
    float* pOut   = out + (size_t)B_ROWS * KSEL + (size_t)row * KSEL;
    idxOut[0] = (float)a;
    pOut[0]   = v;
#pragma unroll
    for (int j = 1; j < KSEL; ++j) {
      const int cand = (j - 1 < a) ? (j - 1) : j;  // smallest indices, skip a
      idxOut[j] = (float)cand;
      pOut[j]   = 0.0f;
    }
    atomicAdd(&ws[0], v);                                        // Σ v
    atomicAdd(&ws[1], v * (LN2f * fastlog2(v + EPSF)));          // Σ v·ln(v+eps)
  }
}

__global__ void finalize_kernel(const float* __restrict__ ws,
                                float* __restrict__ out) {
  const float sv  = ws[0];
  const float svl = ws[1];
  const float sparsity     = 2e-4f * (sv / (float)B_ROWS);
  const float entropy      = -(svl / (float)B_ROWS);
  const float entropy_loss = -1e-3f * entropy;
  float* scal = out + (size_t)2 * B_ROWS * KSEL;
  scal[0] = sparsity;
  scal[1] = entropy_loss;
  scal[2] = sparsity + entropy_loss;
}

extern "C" void kernel_launch(void* const* d_in, const int* in_sizes, int n_in,
                              void* d_out, int out_size, void* d_ws, size_t ws_size,
                              hipStream_t stream) {
  (void)in_sizes; (void)n_in; (void)out_size; (void)ws_size;
  const float* logits = (const float*)d_in[0];
  const float* noise  = (const float*)d_in[1];
  float* out = (float*)d_out;
  float* ws  = (float*)d_ws;

  hipMemsetAsync(d_ws, 0, 2 * sizeof(float), stream);  // capture-safe
  router_kernel<<<dim3(B_ROWS), dim3(THREADS), 0, stream>>>(logits, noise, out, ws);
  finalize_kernel<<<dim3(1), dim3(1), 0, stream>>>(ws, out);
}